// ResidualVectorQuantization_79894981640484
// MI455X (gfx1250) — compile-verified
//
#include <hip/hip_runtime.h>
#include <math.h>

typedef float v2f __attribute__((ext_vector_type(2)));
typedef float v8f __attribute__((ext_vector_type(8)));

#define B_N 8192
#define D_N 512
#define Q_N 4
#define K_N 1024

// ---------------------------------------------------------------------------
// zero scratch scalars
__global__ void zero_kernel(float* p, int n) {
    int i = threadIdx.x;
    if (i < n) p[i] = 0.0f;
}

// ---------------------------------------------------------------------------
// one wave per row: out[row] = sum(src[row]^2); optional atomicAdd(sqrt) for
// the mean-norm term of the diversity denominator.
__global__ __launch_bounds__(256) void rownorm_kernel(
    const float* __restrict__ src, float* __restrict__ out,
    float* __restrict__ normsum) {
    int wave = (blockIdx.x * 256 + threadIdx.x) >> 5;
    int lane = threadIdx.x & 31;
    const float4* r4 = (const float4*)(src + (size_t)wave * D_N);
    float s = 0.0f;
    #pragma unroll
    for (int i = 0; i < 4; ++i) {
        float4 v = r4[lane + 32 * i];
        s += v.x * v.x + v.y * v.y + v.z * v.z + v.w * v.w;
    }
    #pragma unroll
    for (int m = 16; m >= 1; m >>= 1) s += __shfl_xor(s, m, 32);
    if (lane == 0) {
        out[wave] = s;
        if (normsum) atomicAdd(normsum, sqrtf(s));
    }
}

// ---------------------------------------------------------------------------
// Ping-pong software-pipelined 16x128 x K=512 fp32 WMMA GEMM for one wave.
//  A: 16 rows staged in LDS (ldsA, row stride D_N)
//  B: 128 rows of bmat starting at row colBase (streamed from global/L2)
//  c[t]: eight 16x16 fp32 accumulator tiles (register resident)
// Per ISA layouts: A lanes0-15 hold K=d0,d0+1; lanes16-31 K=d0+2,d0+3.
// C: lane = N (mod 16), VGPR v = M (+8 for lanes 16-31).
// 128 k-steps; 2x-unrolled double buffer -> no register rotation copies.
__device__ __forceinline__ void wmma_gemm_row16(
    const float* __restrict__ bmat, int colBase,
    const float* __restrict__ ldsA, int l15, int kh, v8f c[8]) {
    const float* bp[8];
    #pragma unroll
    for (int t = 0; t < 8; ++t)
        bp[t] = bmat + (size_t)(colBase + t * 16 + l15) * D_N + 2 * kh;
    const float* ap = ldsA + l15 * D_N + 2 * kh;

    v2f a0, a1, b0[8], b1[8];

    // prologue: k-step 0
    a0 = *(const v2f*)ap;
    #pragma unroll
    for (int t = 0; t < 8; ++t) b0[t] = *(const v2f*)bp[t];

    // steady state: steps 0..125 consumed here, prefetch runs one step ahead
    for (int s = 0; s < 126; s += 2) {
        const int d1 = 4 * (s + 1), d2 = 4 * (s + 2);
        a1 = *(const v2f*)(ap + d1);
        #pragma unroll
        for (int t = 0; t < 8; ++t) b1[t] = *(const v2f*)(bp[t] + d1);
        #pragma unroll
        for (int t = 0; t < 8; ++t)
            c[t] = __builtin_amdgcn_wmma_f32_16x16x4_f32(
                false, a0, false, b0[t], (short)0, c[t], false, false);
        a0 = *(const v2f*)(ap + d2);
        #pragma unroll
        for (int t = 0; t < 8; ++t) b0[t] = *(const v2f*)(bp[t] + d2);
        #pragma unroll
        for (int t = 0; t < 8; ++t)
            c[t] = __builtin_amdgcn_wmma_f32_16x16x4_f32(
                false, a1, false, b1[t], (short)0, c[t], false, false);
    }
    // epilogue: steps 126 (already in a0/b0) and 127
    a1 = *(const v2f*)(ap + 4 * 127);
    #pragma unroll
    for (int t = 0; t < 8; ++t) b1[t] = *(const v2f*)(bp[t] + 4 * 127);
    #pragma unroll
    for (int t = 0; t < 8; ++t)
        c[t] = __builtin_amdgcn_wmma_f32_16x16x4_f32(
            false, a0, false, b0[t], (short)0, c[t], false, false);
    #pragma unroll
    for (int t = 0; t < 8; ++t)
        c[t] = __builtin_amdgcn_wmma_f32_16x16x4_f32(
            false, a1, false, b1[t], (short)0, c[t], false, false);
}

// ---------------------------------------------------------------------------
// Pairwise-distance sum: block = 16 rows of x (LDS) x 1024 cols (8 waves x
// eight 16x16 WMMA tiles over K=512).
__global__ __launch_bounds__(256) void diversity_kernel(
    const float* __restrict__ x, const float* __restrict__ n2,
    float* __restrict__ distsum) {
    __shared__ float xT[16 * D_N];   // 32 KB
    __shared__ float wacc[8];

    const int iBase = blockIdx.x * 16;
    const int tid = threadIdx.x;
    const int wv = tid >> 5, lane = tid & 31;
    const int l15 = lane & 15, kh = lane >> 4;

    {   // 16 rows are contiguous in memory -> straight 32KB vector copy
        const float4* s4 = (const float4*)(x + (size_t)iBase * D_N);
        float4* d4 = (float4*)xT;
        for (int k = tid; k < 16 * D_N / 4; k += 256) d4[k] = s4[k];
    }
    __syncthreads();

    const int jBase = blockIdx.y * 1024 + wv * 128;

    v8f c[8];
    #pragma unroll
    for (int t = 0; t < 8; ++t)
        #pragma unroll
        for (int v = 0; v < 8; ++v) c[t][v] = 0.0f;

    wmma_gemm_row16(x, jBase, xT, l15, kh, c);

    const int mBase = kh * 8;
    float s = 0.0f;
    #pragma unroll
    for (int t = 0; t < 8; ++t) {
        int n = jBase + t * 16 + l15;
        float w2n = n2[n];
        #pragma unroll
        for (int v = 0; v < 8; ++v) {
            int m = iBase + mBase + v;
            float d2 = n2[m] + w2n - 2.0f * c[t][v];
            float d = (m == n) ? 0.0f : sqrtf(fmaxf(d2, 0.0f));
            s += d;
        }
    }
    #pragma unroll
    for (int m = 16; m >= 1; m >>= 1) s += __shfl_xor(s, m, 32);
    if (lane == 0) wacc[wv] = s;
    __syncthreads();
    if (tid == 0) {
        float t = 0.0f;
        for (int i = 0; i < 8; ++i) t += wacc[i];
        atomicAdd(distsum, t);
    }
}

// ---------------------------------------------------------------------------
// One RVQ stage. Block owns 16 rows; computes full 16x1024 dist^2 with WMMA
// (kept in registers), then argmin / softmax / entropy, then gather+update.
__global__ __launch_bounds__(256) void rvq_stage_kernel(
    const float* __restrict__ src,    // residual in (x for stage 0)
    float* __restrict__ resid,        // residual out
    const float* __restrict__ cb,     // (K,D) codebook for this stage
    const float* __restrict__ w2,     // (K) codeword norms
    float* __restrict__ quant,        // (B,Q,D)
    float* __restrict__ outIdx,       // (B,Q) indices as float
    float* __restrict__ commitAcc,    // scalar: sum (r-q)^2
    float* __restrict__ entAcc,       // scalar: sum of per-row entropies
    int stage) {
    __shared__ float rT[16 * D_N];    // 32 KB residual tile
    __shared__ float r2s[16];
    __shared__ float redA[8 * 16];
    __shared__ int   redI[8 * 16];
    __shared__ float rowMin[16];
    __shared__ int   rowIdx[16];
    __shared__ float rowZ[16];
    __shared__ float wacc[8];

    const int rowBase = blockIdx.x * 16;
    const int tid = threadIdx.x;
    const int wv = tid >> 5, lane = tid & 31;
    const int l15 = lane & 15, kh = lane >> 4;

    {   // load residual tile (contiguous 32KB)
        const float4* s4 = (const float4*)(src + (size_t)rowBase * D_N);
        float4* d4 = (float4*)rT;
        for (int k = tid; k < 16 * D_N / 4; k += 256) d4[k] = s4[k];
    }
    __syncthreads();

    // per-row |r|^2 from LDS (2 rows per wave)
    #pragma unroll
    for (int rr = 0; rr < 2; ++rr) {
        int row = wv + rr * 8;
        float s = 0.0f;
        #pragma unroll
        for (int i = 0; i < 16; ++i) {
            float v = rT[row * D_N + lane + 32 * i];
            s += v * v;
        }
        #pragma unroll
        for (int m = 16; m >= 1; m >>= 1) s += __shfl_xor(s, m, 32);
        if (lane == 0) r2s[row] = s;
    }
    __syncthreads();

    // ---- GEMM: dist2 tiles in registers ----
    const int colBase = wv * 128;
    v8f c[8];
    #pragma unroll
    for (int t = 0; t < 8; ++t)
        #pragma unroll
        for (int v = 0; v < 8; ++v) c[t][v] = 0.0f;

    wmma_gemm_row16(cb, colBase, rT, l15, kh, c);

    const int mBase = kh * 8;
    #pragma unroll
    for (int t = 0; t < 8; ++t) {
        int n = colBase + t * 16 + l15;
        float wn = w2[n];
        #pragma unroll
        for (int v = 0; v < 8; ++v)
            c[t][v] = r2s[mBase + v] + wn - 2.0f * c[t][v];   // dist^2
    }

    // ---- argmin per row (first-occurrence tie-break) ----
    #pragma unroll
    for (int v = 0; v < 8; ++v) {
        float mn = c[0][v];
        int mi = colBase + l15;
        #pragma unroll
        for (int t = 1; t < 8; ++t) {
            float dv = c[t][v];
            int di = colBase + t * 16 + l15;
            if (dv < mn || (dv == mn && di < mi)) { mn = dv; mi = di; }
        }
        #pragma unroll
        for (int msk = 1; msk < 16; msk <<= 1) {  // reduce within half-wave
            float om = __shfl_xor(mn, msk, 32);
            int   oi = __shfl_xor(mi, msk, 32);
            if (om < mn || (om == mn && oi < mi)) { mn = om; mi = oi; }
        }
        if (l15 == 0) { redA[wv * 16 + mBase + v] = mn; redI[wv * 16 + mBase + v] = mi; }
    }
    __syncthreads();
    if (tid < 16) {
        float mn = redA[tid]; int mi = redI[tid];
        for (int w = 1; w < 8; ++w) {
            float om = redA[w * 16 + tid]; int oi = redI[w * 16 + tid];
            if (om < mn || (om == mn && oi < mi)) { mn = om; mi = oi; }
        }
        rowMin[tid] = mn; rowIdx[tid] = mi;
    }
    __syncthreads();

    // ---- softmax partition function Z per row (T=1, shifted by -min) ----
    #pragma unroll
    for (int v = 0; v < 8; ++v) {
        int m = mBase + v;
        float rm = rowMin[m], s = 0.0f;
        #pragma unroll
        for (int t = 0; t < 8; ++t) s += expf(rm - c[t][v]);
        #pragma unroll
        for (int msk = 1; msk < 16; msk <<= 1) s += __shfl_xor(s, msk, 32);
        if (l15 == 0) redA[wv * 16 + m] = s;
    }
    __syncthreads();
    if (tid < 16) {
        float s = 0.0f;
        for (int w = 0; w < 8; ++w) s += redA[w * 16 + tid];
        rowZ[tid] = s;
    }
    __syncthreads();

    // ---- entropy per row: -sum p*log(p+eps) ----
    #pragma unroll
    for (int v = 0; v < 8; ++v) {
        int m = mBase + v;
        float rm = rowMin[m], Z = rowZ[m], s = 0.0f;
        #pragma unroll
        for (int t = 0; t < 8; ++t) {
            float p = expf(rm - c[t][v]) / Z;
            s += p * logf(p + 1e-8f);
        }
        #pragma unroll
        for (int msk = 1; msk < 16; msk <<= 1) s += __shfl_xor(s, msk, 32);
        if (l15 == 0) redA[wv * 16 + m] = s;
    }
    __syncthreads();
    if (tid < 16) {
        float s = 0.0f;
        for (int w = 0; w < 8; ++w) s += redA[w * 16 + tid];
        atomicAdd(entAcc, -s);
        outIdx[(size_t)(rowBase + tid) * Q_N + stage] = (float)rowIdx[tid];
    }
    __syncthreads();

    // ---- gather q, write quantized (== q_ste value), update residual ----
    // float4 (b128) path: 2048 float4 elements / 256 threads = 8 each.
    float cacc = 0.0f;
    #pragma unroll
    for (int k = 0; k < 8; ++k) {
        int e4 = tid + 256 * k;          // float4 index within 16x512 tile
        int row = e4 >> 7;               // 128 float4 per row
        int col4 = e4 & 127;
        float4 q4 = *(const float4*)(cb + (size_t)rowIdx[row] * D_N + col4 * 4);
        float4 r4 = ((const float4*)rT)[e4];
        float4 df;
        df.x = r4.x - q4.x; df.y = r4.y - q4.y;
        df.z = r4.z - q4.z; df.w = r4.w - q4.w;
        cacc += df.x * df.x + df.y * df.y + df.z * df.z + df.w * df.w;
        *(float4*)(quant + ((size_t)(rowBase + row) * Q_N + stage) * D_N + col4 * 4) = q4;
        ((float4*)(resid + (size_t)rowBase * D_N))[e4] = df;
    }
    #pragma unroll
    for (int m = 16; m >= 1; m >>= 1) cacc += __shfl_xor(cacc, m, 32);
    if (lane == 0) wacc[wv] = cacc;
    __syncthreads();
    if (tid == 0) {
        float s = 0.0f;
        for (int i = 0; i < 8; ++i) s += wacc[i];
        atomicAdd(commitAcc, s);   // for last stage this IS sum(resid_final^2)
    }
}

// ---------------------------------------------------------------------------
// scal: [0]=dist_sum [1]=norm_sum [2..5]=commitRaw [6..9]=entropy*B
__global__ void finalize_kernel(const float* __restrict__ scal,
                                float* __restrict__ loss) {
    const float Bf = (float)B_N;
    const float BD = (float)(B_N * D_N);
    float mean_dist = scal[0] / (Bf * (Bf - 1.0f));
    float denom = scal[1] / Bf + 1e-8f;
    float divn = fminf(fmaxf(mean_dist / denom, 0.0f), 1.0f);

    float total_commit = 0.0f, total_entropy = 0.0f;
    for (int i = 0; i < Q_N; ++i) {
        float commit = 1.25f * scal[2 + i] / BD;   // (1 + BETA_COMMIT)*mean(diff^2)
        float entropy = scal[6 + i] / Bf;
        float perp = expf(entropy);
        float commit_norm = 1.0f / (1.0f + expf(-10.0f * commit));
        float perp_norm = perp / (float)K_N;
        float aw = 1.0f * (2.0f - divn) / (1.0f + commit_norm * perp_norm);
        total_commit += commit;
        total_entropy -= aw * entropy;
    }
    float res_loss = 0.5f * scal[2 + 3] / BD;   // LAMBDA_RES * mean(resid^2)
    *loss = res_loss + 0.25f * total_commit / (float)Q_N +
            total_entropy / (float)Q_N;
}

// ---------------------------------------------------------------------------
extern "C" void kernel_launch(void* const* d_in, const int* in_sizes, int n_in,
                              void* d_out, int out_size, void* d_ws, size_t ws_size,
                              hipStream_t stream) {
    const float* x   = (const float*)d_in[0];   // (B,D)
    const float* cbs = (const float*)d_in[1];   // (Q,K,D)

    float* out    = (float*)d_out;
    float* quant  = out;                                 // B*Q*D
    float* outIdx = out + (size_t)B_N * Q_N * D_N;       // B*Q (as float)
    float* loss   = outIdx + (size_t)B_N * Q_N;          // 1

    float* ws    = (float*)d_ws;
    float* resid = ws;                                   // B*D
    float* n2x   = ws + (size_t)B_N * D_N;               // B
    float* w2    = n2x + B_N;                            // Q*K
    float* scal  = w2 + Q_N * K_N;                       // 16 scalars

    zero_kernel<<<1, 32, 0, stream>>>(scal, 16);
    rownorm_kernel<<<B_N / 8, 256, 0, stream>>>(x, n2x, scal + 1);
    rownorm_kernel<<<(Q_N * K_N) / 8, 256, 0, stream>>>(cbs, w2, nullptr);

    dim3 dg(B_N / 16, B_N / 1024);
    diversity_kernel<<<dg, 256, 0, stream>>>(x, n2x, scal + 0);

    for (int s = 0; s < Q_N; ++s) {
        rvq_stage_kernel<<<B_N / 16, 256, 0, stream>>>(
            s == 0 ? x : resid, resid,
            cbs + (size_t)s * K_N * D_N, w2 + (size_t)s * K_N,
            quant, outIdx, scal + 2 + s, scal + 6 + s, s);
    }
    finalize_kernel<<<1, 1, 0, stream>>>(scal, loss);
}